// DynamicsSolver_30202210026033
// MI455X (gfx1250) — compile-verified
//
#include <hip/hip_runtime.h>
#include <math.h>

// ---------------------------------------------------------------------------
// DynamicsSolver fused CDNA5 implementation (gfx1250, wave32, WMMA f32).
//
// Structure:
//   K1 node_kernel : node_enc MLP+LN -> latent[N,128]; proc-block per-node
//                    heads nw/m/i/fext/text (only proc's survive to output).
//   K2 edge_kernel : fully fused per-edge pipeline (frame calc, init+proc
//                    encoders, inter MLPs, residual LN chain il1->il2->il3,
//                    decoders i1/i2/fsc, fij/tau, atomic segment-sum).
//                    Exploits: decoders of non-final blocks are dead code,
//                    and inter_proc(pre) is identical for both proc passes.
//   K3 finalize    : ri_lin/ri_ang + cor_v/cor_w tiny MLPs -> out[N,10].
// ---------------------------------------------------------------------------

#define LATENT 128

typedef __attribute__((ext_vector_type(2))) float v2f;
typedef __attribute__((ext_vector_type(8))) float v8f;

#ifndef __has_builtin
#define __has_builtin(x) 0
#endif
#if __has_builtin(__builtin_amdgcn_wmma_f32_16x16x4_f32)
#define HAVE_WMMA4 1
#else
#define HAVE_WMMA4 0
#endif

struct Mlp { const float *W1, *b1, *W2, *b2, *g, *be; };

struct F3 { float x, y, z; };
__device__ __forceinline__ F3 operator+(F3 a, F3 b) { return {a.x+b.x, a.y+b.y, a.z+b.z}; }
__device__ __forceinline__ F3 operator-(F3 a, F3 b) { return {a.x-b.x, a.y-b.y, a.z-b.z}; }
__device__ __forceinline__ F3 operator*(F3 a, float s) { return {a.x*s, a.y*s, a.z*s}; }
__device__ __forceinline__ float dot3(F3 a, F3 b) { return a.x*b.x + a.y*b.y + a.z*b.z; }
__device__ __forceinline__ F3 cross3(F3 a, F3 b) {
  return {a.y*b.z - a.z*b.y, a.z*b.x - a.x*b.z, a.x*b.y - a.y*b.x};
}
__device__ __forceinline__ F3 nrm3(F3 a) {  // t / clip(|t|, 1e-6)
  float n = fmaxf(sqrtf(dot3(a, a)), 1e-6f);
  return a * (1.f / n);
}
__device__ __forceinline__ F3 ld3(const float* p, int i) { return {p[3*i], p[3*i+1], p[3*i+2]}; }

// ---------------------------------------------------------------------------
// 16xLATENT GEMM: O[16][128] = act(A[16][K] * W[K][128] + bias), one wave.
// A in LDS (row-major, stride LDA), W row-major in global (L2-resident).
// Uses V_WMMA_F32_16X16X4_F32: A frag lane l = {A[l&15][k0+2*(l>>4)], +1},
// B frag lane l = {W[k0+2*(l>>4)][n0+(l&15)], next k row}, D VGPR v at lane l
// = elem (M = v + 8*(l>>4), N = l&15).
// ---------------------------------------------------------------------------
template <int K, int LDA, bool RELU>
__device__ __forceinline__ void gemm16(const float* __restrict__ A,
                                       const float* __restrict__ W,
                                       const float* __restrict__ bias,
                                       float* __restrict__ O) {
  const int lane = threadIdx.x & 31;
  const int row  = lane & 15;        // A row / D column
  const int hi   = lane >> 4;
  const int koff = hi << 1;
#pragma unroll
  for (int nt = 0; nt < 8; ++nt) {
    const int n0 = nt << 4;
    v8f acc = {0.f, 0.f, 0.f, 0.f, 0.f, 0.f, 0.f, 0.f};
    for (int k0 = 0; k0 < K; k0 += 4) {
#if HAVE_WMMA4
      const int ka = k0 + koff;
      v2f a, b;
      a.x = (ka     < K) ? A[row * LDA + ka]     : 0.f;
      a.y = (ka + 1 < K) ? A[row * LDA + ka + 1] : 0.f;
      b.x = (ka     < K) ? W[ka * LATENT + n0 + row]       : 0.f;
      b.y = (ka + 1 < K) ? W[(ka + 1) * LATENT + n0 + row] : 0.f;
      acc = __builtin_amdgcn_wmma_f32_16x16x4_f32(false, a, false, b,
                                                  (short)0, acc, false, false);
#else
      // VALU fallback replicating the WMMA D layout (compile-safety net).
#pragma unroll
      for (int v = 0; v < 8; ++v) {
        const int m = v + (hi << 3);
#pragma unroll
        for (int kk = 0; kk < 4; ++kk) {
          const int k = k0 + kk;
          if (k < K) acc[v] += A[m * LDA + k] * W[k * LATENT + n0 + row];
        }
      }
#endif
    }
#pragma unroll
    for (int v = 0; v < 8; ++v) {
      const int m = v + (hi << 3);
      float val = acc[v] + bias[n0 + row];
      if (RELU) val = fmaxf(val, 0.f);
      O[m * LATENT + n0 + row] = val;
    }
  }
}

// Row-wise LayerNorm over 128 features of a 16x128 LDS tile (one wave:
// two half-lanes per row, combined with a cross-half shuffle).
__device__ __forceinline__ void layernorm16(float* __restrict__ Y,
                                            const float* __restrict__ g,
                                            const float* __restrict__ b) {
  const int lane = threadIdx.x & 31;
  const int row  = lane & 15;
  const int c0   = (lane >> 4) << 6;
  float s = 0.f, ss = 0.f;
#pragma unroll 8
  for (int c = 0; c < 64; ++c) {
    const float v = Y[row * LATENT + c0 + c];
    s += v; ss += v * v;
  }
  s  += __shfl_xor(s, 16, 32);
  ss += __shfl_xor(ss, 16, 32);
  const float mu  = s * (1.f / 128.f);
  const float var = ss * (1.f / 128.f) - mu * mu;
  const float inv = rsqrtf(var + 1e-5f);
#pragma unroll 8
  for (int c = 0; c < 64; ++c) {
    const int cc = c0 + c;
    Y[row * LATENT + cc] = (Y[row * LATENT + cc] - mu) * inv * g[cc] + b[cc];
  }
}

// Small output head: O[16][DOUT] = H[16][128] @ W2[128][DOUT] + b2.
template <int DOUT>
__device__ __forceinline__ void head16(const float* __restrict__ H,
                                       const float* __restrict__ W2,
                                       const float* __restrict__ b2,
                                       float* __restrict__ O) {
  const int lane = threadIdx.x & 31;
  const int row  = lane & 15;
  const int c0   = (lane >> 4) << 6;
#pragma unroll
  for (int o = 0; o < DOUT; ++o) {
    float s = 0.f;
    for (int c = 0; c < 64; ++c)
      s += H[row * LATENT + c0 + c] * W2[(c0 + c) * DOUT + o];
    s += __shfl_xor(s, 16, 32);
    if (lane < 16) O[row * DOUT + o] = s + b2[o];
  }
}

// ---------------------------------------------------------------------------
// Kernel 1: node encoder + per-node proc heads.
// ---------------------------------------------------------------------------
struct NodeParams { Mlp enc, nw, m, inert, fext, text; };

__global__ void __launch_bounds__(32)
node_kernel(const float* __restrict__ sfeat, int N, NodeParams P,
            float* __restrict__ latent, float* __restrict__ nw_p,
            float* __restrict__ m_p, float* __restrict__ i_p,
            float* __restrict__ fext_p, float* __restrict__ text_p) {
  __shared__ float X[16 * 16];
  __shared__ float H[16 * LATENT];
  __shared__ float L[16 * LATENT];
  __shared__ float O[16 * 3];
  const int lane = threadIdx.x;
  const int base = blockIdx.x * 16;

  for (int i = lane; i < 256; i += 32) {
    int nr = base + (i >> 4); if (nr >= N) nr = N - 1;
    X[i] = sfeat[nr * 16 + (i & 15)];
  }
  __syncthreads();
  gemm16<16, 16, true>(X, P.enc.W1, P.enc.b1, H);        __syncthreads();
  gemm16<128, LATENT, false>(H, P.enc.W2, P.enc.b2, L);  __syncthreads();
  layernorm16(L, P.enc.g, P.enc.be);                     __syncthreads();
  for (int i = lane; i < 16 * LATENT; i += 32) {
    const int nr = base + (i >> 7);
    if (nr < N) latent[nr * LATENT + (i & 127)] = L[i];
  }
  __syncthreads();

  // proc-block heads (the only per-node heads that reach the output)
  gemm16<128, LATENT, true>(L, P.nw.W1, P.nw.b1, H);     __syncthreads();
  head16<1>(H, P.nw.W2, P.nw.b2, O);                     __syncthreads();
  if (lane < 16 && base + lane < N) nw_p[base + lane] = O[lane];
  __syncthreads();
  gemm16<128, LATENT, true>(L, P.m.W1, P.m.b1, H);       __syncthreads();
  head16<1>(H, P.m.W2, P.m.b2, O);                       __syncthreads();
  if (lane < 16 && base + lane < N) m_p[base + lane] = O[lane];
  __syncthreads();
  gemm16<128, LATENT, true>(L, P.inert.W1, P.inert.b1, H); __syncthreads();
  head16<1>(H, P.inert.W2, P.inert.b2, O);               __syncthreads();
  if (lane < 16 && base + lane < N) i_p[base + lane] = O[lane];
  __syncthreads();
  gemm16<128, LATENT, true>(L, P.fext.W1, P.fext.b1, H); __syncthreads();
  head16<3>(H, P.fext.W2, P.fext.b2, O);                 __syncthreads();
  if (lane < 16 && base + lane < N)
    for (int o = 0; o < 3; ++o) fext_p[(base + lane) * 3 + o] = O[lane * 3 + o];
  __syncthreads();
  gemm16<128, LATENT, true>(L, P.text.W1, P.text.b1, H); __syncthreads();
  head16<3>(H, P.text.W2, P.text.b2, O);                 __syncthreads();
  if (lane < 16 && base + lane < N)
    for (int o = 0; o < 3; ++o) text_p[(base + lane) * 3 + o] = O[lane * 3 + o];
}

// ---------------------------------------------------------------------------
// Kernel 2: fused edge pipeline.
// ---------------------------------------------------------------------------
struct EdgeParams {
  Mlp nf_i, ef_i, in_i;      // init block encoders
  Mlp nf_p, ef_p, in_p;      // proc block encoders
  Mlp i1, i2, fsc;           // proc decoders (last block only)
  const float *ln_g, *ln_b;  // proc residual LayerNorm
  const float *v_scale, *dx_min, *dx_max;
};

__global__ void __launch_bounds__(32)
edge_kernel(const int* __restrict__ ei, int E,
            const float* __restrict__ pos, const float* __restrict__ vt,
            const float* __restrict__ vtm1, const float* __restrict__ wvec,
            const float* __restrict__ eattr,
            const float* __restrict__ latent, const float* __restrict__ nw_p,
            float* __restrict__ out_f, float* __restrict__ out_t,
            EdgeParams P) {
  __shared__ float AI[16 * 384];        // inter input [spl+rpl | nl_sum | el]
  __shared__ float H[16 * LATENT];
  __shared__ float Y[16 * LATENT];
  __shared__ float IP[16 * LATENT];     // LN(inter_proc(pre)) (reused twice)
  __shared__ float LT[16 * LATENT];     // running latent il_k
  __shared__ float SF[16 * 8], RF[16 * 8], EF[16 * 8];
  __shared__ float GVA[16 * 3], GVB[16 * 3], GVC[16 * 3];
  __shared__ float SPs[16 * 3], RPs[16 * 3];
  __shared__ float WSR[16 * 2];
  __shared__ float CF[16 * 3], CA[16 * 3], LAM[16];
  __shared__ int RIX[16];

  const int lane = threadIdx.x;
  const int tile = blockIdx.x;

  if (lane < 16) {                      // per-edge geometry, one lane per edge
    const int eg = tile * 16 + lane;
    const int e  = (eg < E) ? eg : E - 1;
    const int s = ei[e], r = ei[E + e];
    F3 sp = ld3(pos, s),  rp = ld3(pos, r);
    F3 sv = ld3(vt, s),   rv = ld3(vt, r);
    F3 svm = ld3(vtm1, s), rvm = ld3(vtm1, r);
    F3 sw = ld3(wvec, s), rw = ld3(wvec, r);
    F3 rel = rp - sp;
    const float dist = sqrtf(dot3(rel, rel));
    F3 va = rel * (1.f / dist);
    F3 b = nrm3(cross3(rv - sv, va)) + nrm3(sv + rv)
         + nrm3(cross3(rw - sw, va)) + nrm3(sw + rw)
         + nrm3(cross3(rvm - svm, va)) + nrm3(svm + rvm);
    b.x += 1.f; b.y += 1.f; b.z += 1.f;
    F3 b_prl = va * dot3(b, va);
    F3 b_prp = b - b_prl;
    F3 vb = nrm3(cross3(b_prp, va));
    F3 vc = nrm3(cross3(b_prl, vb));
    const float inv_vs = 1.f / (*P.v_scale);
    SF[lane*8+0] =  dot3(va, sv) * inv_vs;
    SF[lane*8+1] =  dot3(vb, sv) * inv_vs;
    SF[lane*8+2] =  dot3(vc, sv) * inv_vs;
    SF[lane*8+3] =  dot3(va, sw);
    SF[lane*8+4] =  dot3(vb, sw);
    SF[lane*8+5] =  dot3(vc, sw);
    SF[lane*8+6] = 0.f; SF[lane*8+7] = 0.f;
    RF[lane*8+0] = -dot3(va, rv) * inv_vs;
    RF[lane*8+1] = -dot3(vb, rv) * inv_vs;
    RF[lane*8+2] = -dot3(vc, rv) * inv_vs;
    RF[lane*8+3] = -dot3(va, rw);
    RF[lane*8+4] = -dot3(vb, rw);
    RF[lane*8+5] = -dot3(vc, rw);
    RF[lane*8+6] = 0.f; RF[lane*8+7] = 0.f;
    const float dmin = *P.dx_min, dmax = *P.dx_max;
    EF[lane*8+0] = fabsf((dist - dmin) / (dmax - dmin));  // |edge_dx_| (unit dir)
    EF[lane*8+1] = eattr[e*4+0]; EF[lane*8+2] = eattr[e*4+1];
    EF[lane*8+3] = eattr[e*4+2]; EF[lane*8+4] = eattr[e*4+3];
    EF[lane*8+5] = 0.f; EF[lane*8+6] = 0.f; EF[lane*8+7] = 0.f;
    GVA[lane*3+0]=va.x; GVA[lane*3+1]=va.y; GVA[lane*3+2]=va.z;
    GVB[lane*3+0]=vb.x; GVB[lane*3+1]=vb.y; GVB[lane*3+2]=vb.z;
    GVC[lane*3+0]=vc.x; GVC[lane*3+1]=vc.y; GVC[lane*3+2]=vc.z;
    SPs[lane*3+0]=sp.x; SPs[lane*3+1]=sp.y; SPs[lane*3+2]=sp.z;
    RPs[lane*3+0]=rp.x; RPs[lane*3+1]=rp.y; RPs[lane*3+2]=rp.z;
    WSR[lane*2+0] = nw_p[s]; WSR[lane*2+1] = nw_p[r];
    RIX[lane] = r;
  }
  // nl_sum gather -> AI columns [128,256) (node_latent is L2-resident)
  for (int rr = 0; rr < 16; ++rr) {
    const int eg = tile * 16 + rr;
    const int e  = (eg < E) ? eg : E - 1;
    const int s = ei[e], r = ei[E + e];
    for (int c = lane; c < LATENT; c += 32)
      AI[rr * 384 + 128 + c] = latent[s * LATENT + c] + latent[r * LATENT + c];
  }
  __syncthreads();

  // ===== init encoders -> il1 =====
  gemm16<6, 8, true>(SF, P.nf_i.W1, P.nf_i.b1, H);             __syncthreads();
  gemm16<128, LATENT, false>(H, P.nf_i.W2, P.nf_i.b2, Y);      __syncthreads();
  layernorm16(Y, P.nf_i.g, P.nf_i.be);                         __syncthreads();
  for (int i = lane; i < 16 * LATENT; i += 32) AI[(i >> 7) * 384 + (i & 127)] = Y[i];
  __syncthreads();
  gemm16<6, 8, true>(RF, P.nf_i.W1, P.nf_i.b1, H);             __syncthreads();
  gemm16<128, LATENT, false>(H, P.nf_i.W2, P.nf_i.b2, Y);      __syncthreads();
  layernorm16(Y, P.nf_i.g, P.nf_i.be);                         __syncthreads();
  for (int i = lane; i < 16 * LATENT; i += 32) AI[(i >> 7) * 384 + (i & 127)] += Y[i];
  __syncthreads();
  gemm16<5, 8, true>(EF, P.ef_i.W1, P.ef_i.b1, H);             __syncthreads();
  gemm16<128, LATENT, false>(H, P.ef_i.W2, P.ef_i.b2, Y);      __syncthreads();
  layernorm16(Y, P.ef_i.g, P.ef_i.be);                         __syncthreads();
  for (int i = lane; i < 16 * LATENT; i += 32) AI[(i >> 7) * 384 + 256 + (i & 127)] = Y[i];
  __syncthreads();
  gemm16<384, 384, true>(AI, P.in_i.W1, P.in_i.b1, H);         __syncthreads();
  gemm16<128, LATENT, false>(H, P.in_i.W2, P.in_i.b2, LT);     __syncthreads();
  layernorm16(LT, P.in_i.g, P.in_i.be);                        __syncthreads();

  // ===== proc encoders -> ip (identical for both proc iterations) =====
  gemm16<6, 8, true>(SF, P.nf_p.W1, P.nf_p.b1, H);             __syncthreads();
  gemm16<128, LATENT, false>(H, P.nf_p.W2, P.nf_p.b2, Y);      __syncthreads();
  layernorm16(Y, P.nf_p.g, P.nf_p.be);                         __syncthreads();
  for (int i = lane; i < 16 * LATENT; i += 32) AI[(i >> 7) * 384 + (i & 127)] = Y[i];
  __syncthreads();
  gemm16<6, 8, true>(RF, P.nf_p.W1, P.nf_p.b1, H);             __syncthreads();
  gemm16<128, LATENT, false>(H, P.nf_p.W2, P.nf_p.b2, Y);      __syncthreads();
  layernorm16(Y, P.nf_p.g, P.nf_p.be);                         __syncthreads();
  for (int i = lane; i < 16 * LATENT; i += 32) AI[(i >> 7) * 384 + (i & 127)] += Y[i];
  __syncthreads();
  gemm16<5, 8, true>(EF, P.ef_p.W1, P.ef_p.b1, H);             __syncthreads();
  gemm16<128, LATENT, false>(H, P.ef_p.W2, P.ef_p.b2, Y);      __syncthreads();
  layernorm16(Y, P.ef_p.g, P.ef_p.be);                         __syncthreads();
  for (int i = lane; i < 16 * LATENT; i += 32) AI[(i >> 7) * 384 + 256 + (i & 127)] = Y[i];
  __syncthreads();
  gemm16<384, 384, true>(AI, P.in_p.W1, P.in_p.b1, H);         __syncthreads();
  gemm16<128, LATENT, false>(H, P.in_p.W2, P.in_p.b2, IP);     __syncthreads();
  layernorm16(IP, P.in_p.g, P.in_p.be);                        __syncthreads();

  // ===== residual chain: il2 = LN(ip+il1); il3 = LN(ip+il2) =====
#pragma unroll
  for (int it = 0; it < 2; ++it) {
    for (int i = lane; i < 16 * LATENT; i += 32) LT[i] = IP[i] + LT[i];
    __syncthreads();
    layernorm16(LT, P.ln_g, P.ln_b);
    __syncthreads();
  }

  // ===== decoders (only the final block's output survives) =====
  gemm16<128, LATENT, true>(LT, P.i1.W1, P.i1.b1, H);          __syncthreads();
  head16<3>(H, P.i1.W2, P.i1.b2, CF);                          __syncthreads();
  gemm16<128, LATENT, true>(LT, P.i2.W1, P.i2.b1, H);          __syncthreads();
  head16<3>(H, P.i2.W2, P.i2.b2, CA);                          __syncthreads();
  gemm16<128, LATENT, true>(LT, P.fsc.W1, P.fsc.b1, H);        __syncthreads();
  head16<1>(H, P.fsc.W2, P.fsc.b2, LAM);                       __syncthreads();

  if (lane < 16 && tile * 16 + lane < E) {
    F3 va = {GVA[lane*3], GVA[lane*3+1], GVA[lane*3+2]};
    F3 vb = {GVB[lane*3], GVB[lane*3+1], GVB[lane*3+2]};
    F3 vc = {GVC[lane*3], GVC[lane*3+1], GVC[lane*3+2]};
    F3 fij = va * CF[lane*3] + vb * CF[lane*3+1] + vc * CF[lane*3+2];
    F3 aij = va * CA[lane*3] + vb * CA[lane*3+1] + vc * CA[lane*3+2];
    const float ws = WSR[lane*2], wr = WSR[lane*2+1];
    F3 sp = {SPs[lane*3], SPs[lane*3+1], SPs[lane*3+2]};
    F3 rp = {RPs[lane*3], RPs[lane*3+1], RPs[lane*3+2]};
    F3 r0 = (sp * ws + rp * wr) * (1.f / (ws + wr));
    F3 tau = aij - cross3(rp - r0, fij * LAM[lane]);
    const int r = RIX[lane];
    atomicAdd(&out_f[3*r+0], fij.x);
    atomicAdd(&out_f[3*r+1], fij.y);
    atomicAdd(&out_f[3*r+2], fij.z);
    atomicAdd(&out_t[3*r+0], tau.x);
    atomicAdd(&out_t[3*r+1], tau.y);
    atomicAdd(&out_t[3*r+2], tau.z);
  }
}

// ---------------------------------------------------------------------------
// Kernel 3: residuals + corrector MLPs -> out[N,10]
// ---------------------------------------------------------------------------
__global__ void finalize_kernel(int N,
                                const float* __restrict__ a_t1,
                                const float* __restrict__ al_t1,
                                const float* __restrict__ m_p,
                                const float* __restrict__ i_p,
                                const float* __restrict__ fext_p,
                                const float* __restrict__ text_p,
                                const float* __restrict__ out_f,
                                const float* __restrict__ out_t,
                                Mlp cv, Mlp cw, float* __restrict__ out) {
  const int n = blockIdx.x * blockDim.x + threadIdx.x;
  if (n >= N) return;
  float rl[3], ra[3];
  const float mm = m_p[n], ii = i_p[n];
#pragma unroll
  for (int k = 0; k < 3; ++k) {
    rl[k] = mm * a_t1[3*n+k] - out_f[3*n+k] - fext_p[3*n+k];
    ra[k] = ii * al_t1[3*n+k] - out_t[3*n+k] - text_p[3*n+k];
  }
  const float x0 = sqrtf(rl[0]*rl[0] + rl[1]*rl[1] + rl[2]*rl[2]);
  const float x1 = sqrtf(ra[0]*ra[0] + ra[1]*ra[1] + ra[2]*ra[2]);
  float o[4];
#pragma unroll
  for (int q = 0; q < 2; ++q) {
    const Mlp& M = q ? cw : cv;
    float a0 = M.b2[0], a1 = M.b2[1];
    for (int j = 0; j < 128; ++j) {
      float h = M.b1[j] + x0 * M.W1[j] + x1 * M.W1[128 + j] + M.W1[256 + j];
      h = fmaxf(h, 0.f);
      a0 += h * M.W2[2*j];
      a1 += h * M.W2[2*j + 1];
    }
    o[2*q] = a0; o[2*q+1] = a1;
  }
  float* dst = out + (size_t)n * 10;
  dst[0]=rl[0]; dst[1]=rl[1]; dst[2]=rl[2];
  dst[3]=ra[0]; dst[4]=ra[1]; dst[5]=ra[2];
  dst[6]=o[0];  dst[7]=o[1];  dst[8]=o[2]; dst[9]=o[3];
}

// ---------------------------------------------------------------------------
// Host launch. Params pytree flattened in jax order (dict keys sorted, DFS):
//   d_in[0..8]  : edge_index, pos, v_t, v_tm1, w, scalar_feat, edge_attr,
//                 a_t1, alpha_t1
//   then (relative to p = d_in+9):
//   0..3   cor_v {W1,W2,b1,b2}       4..7   cor_w
//   8..59  init block (edge_feat6, fext4, fsc4, i4, i1:4, i2:4, inter6,
//                      ln_b, ln_g, m4, node_feat6, nw4, text4)
//   60..65 node_enc {W1,W2,b1,b2,be,g}
//   66..117 proc block (same layout as init)
//   118 dx_max, 119 dx_min, 120 v_scale
// ---------------------------------------------------------------------------
extern "C" void kernel_launch(void* const* d_in, const int* in_sizes, int n_in,
                              void* d_out, int out_size, void* d_ws, size_t ws_size,
                              hipStream_t stream) {
  const int* ei      = (const int*)d_in[0];
  const int E        = in_sizes[0] / 2;
  const float* pos   = (const float*)d_in[1];
  const int N        = in_sizes[1] / 3;
  const float* vt    = (const float*)d_in[2];
  const float* vtm1  = (const float*)d_in[3];
  const float* wvec  = (const float*)d_in[4];
  const float* sfeat = (const float*)d_in[5];
  const float* eattr = (const float*)d_in[6];
  const float* a_t1  = (const float*)d_in[7];
  const float* al_t1 = (const float*)d_in[8];
  const float* const* p = (const float* const*)(d_in + 9);

  auto M6 = [&](int b) -> Mlp {  // {W1,W2,b1,b2,be,g} flat -> struct
    return { p[b], p[b+2], p[b+1], p[b+3], p[b+5], p[b+4] };
  };
  auto M4 = [&](int b) -> Mlp {  // {W1,W2,b1,b2}
    return { p[b], p[b+2], p[b+1], p[b+3], nullptr, nullptr };
  };
  const int IB = 8, PB = 66;  // init / proc block bases

  NodeParams NP;
  NP.enc   = M6(60);
  NP.nw    = M4(PB + 44);
  NP.m     = M4(PB + 34);
  NP.inert = M4(PB + 14);
  NP.fext  = M4(PB + 6);
  NP.text  = M4(PB + 48);

  EdgeParams EP;
  EP.nf_i = M6(IB + 38); EP.ef_i = M6(IB + 0); EP.in_i = M6(IB + 26);
  EP.nf_p = M6(PB + 38); EP.ef_p = M6(PB + 0); EP.in_p = M6(PB + 26);
  EP.i1 = M4(PB + 18); EP.i2 = M4(PB + 22); EP.fsc = M4(PB + 10);
  EP.ln_g = p[PB + 33]; EP.ln_b = p[PB + 32];
  EP.dx_max = p[118]; EP.dx_min = p[119]; EP.v_scale = p[120];

  Mlp cv = M4(0), cw = M4(4);

  // workspace layout (floats)
  float* ws      = (float*)d_ws;
  float* latent  = ws;                              // N*128
  float* nw_p    = latent + (size_t)N * LATENT;     // N
  float* m_p     = nw_p + N;                        // N
  float* i_p     = m_p + N;                         // N
  float* fext_p  = i_p + N;                         // 3N
  float* text_p  = fext_p + (size_t)3 * N;          // 3N
  float* out_f   = text_p + (size_t)3 * N;          // 3N (zeroed)
  float* out_t   = out_f + (size_t)3 * N;           // 3N (zeroed)

  hipMemsetAsync(out_f, 0, sizeof(float) * 6 * (size_t)N, stream);

  node_kernel<<<(N + 15) / 16, 32, 0, stream>>>(
      sfeat, N, NP, latent, nw_p, m_p, i_p, fext_p, text_p);

  edge_kernel<<<(E + 15) / 16, 32, 0, stream>>>(
      ei, E, pos, vt, vtm1, wvec, eattr, latent, nw_p, out_f, out_t, EP);

  finalize_kernel<<<(N + 255) / 256, 256, 0, stream>>>(
      N, a_t1, al_t1, m_p, i_p, fext_p, text_p, out_f, out_t, cv, cw,
      (float*)d_out);
}